// SimpleALSHAttention16_15650860826845
// MI455X (gfx1250) — compile-verified
//
#include <hip/hip_runtime.h>

typedef __attribute__((ext_vector_type(2))) float v2f;
typedef __attribute__((ext_vector_type(8))) float v8f;

#define SDIM   2048
#define DDIM   64
#define ADIM   66      // D + 2
#define KDIM   68      // padded to multiple of 4
#define ROWS   16      // t-rows per workgroup (one WMMA M-tile)
#define NWAVES 8

// monotone float->uint key: larger float <=> larger unsigned (NaNs pre-cleaned)
__device__ __forceinline__ unsigned mkey(unsigned b) {
  return b ^ ((unsigned)((int)b >> 31) | 0x80000000u);
}

// ---------------------------------------------------------------------------
// Kernel 1: global max of per-row sum-of-squares of qk (monotone float bits)
// ---------------------------------------------------------------------------
__global__ __launch_bounds__(256) void k_maxnorm(const float* __restrict__ qk,
                                                 unsigned* __restrict__ wsmax,
                                                 int nrows) {
  int row  = blockIdx.x * NWAVES + (threadIdx.x >> 5);
  int lane = threadIdx.x & 31;
  if (row >= nrows) return;
  const float* r = qk + (size_t)row * DDIM;
  float v0 = r[lane], v1 = r[lane + 32];
  float s = v0 * v0 + v1 * v1;
  #pragma unroll
  for (int off = 16; off; off >>= 1) s += __shfl_xor(s, off, 32);
  if (lane == 0) atomicMax(wsmax, __float_as_uint(s));  // s >= 0: bits monotone
}

// ---------------------------------------------------------------------------
// Kernel 2: build Pq[row, 0..67] = Q[row] * p[row]  (p = [qk/M, 0, c], pad 2)
// ---------------------------------------------------------------------------
__global__ __launch_bounds__(256) void k_build_pq(const float* __restrict__ qk,
                                                  const float* __restrict__ a,
                                                  const unsigned* __restrict__ wsmax,
                                                  float* __restrict__ pq,
                                                  int nrows) {
  int row  = blockIdx.x * NWAVES + (threadIdx.x >> 5);
  int lane = threadIdx.x & 31;
  if (row >= nrows) return;
  float M = sqrtf(__uint_as_float(*wsmax));
  const float* qr = qk + (size_t)row * DDIM;
  const float* ar = a  + (size_t)row * ADIM;
  float q0 = qr[lane]      / M;
  float q1 = qr[lane + 32] / M;
  float ss  = q0 * q0 + q1 * q1;
  float dot = q0 * ar[lane] + q1 * ar[lane + 32];
  #pragma unroll
  for (int off = 16; off; off >>= 1) {
    ss  += __shfl_xor(ss,  off, 32);
    dot += __shfl_xor(dot, off, 32);
  }
  float c = sqrtf(1.0f - ss);          // NaN possible at the max row (faithful)
  float Q = dot + c * ar[DDIM];        // <q, a[row]> with q = [qk/M, c, 0]
  float* pr = pq + (size_t)row * KDIM;
  pr[lane]      = Q * q0;
  pr[lane + 32] = Q * q1;
  if (lane == 0) { pr[64] = 0.0f; pr[65] = Q * c; pr[66] = 0.0f; pr[67] = 0.0f; }
}

// ---------------------------------------------------------------------------
// Kernel 3: fused  scores = A @ Pq^T (fp32 WMMA 16x16x4) -> radix top-k -> mask
// One workgroup (8 waves) per 16-row t-tile of one (b,h) slice.
// ---------------------------------------------------------------------------
__global__ __launch_bounds__(256) void k_scores_topk(const float* __restrict__ a,
                                                     const float* __restrict__ pq,
                                                     const int*   __restrict__ bucket_p,
                                                     float* __restrict__ out) {
  __shared__ float    Alds[ROWS * KDIM];        //  4.25 KB
  __shared__ float    Sc[ROWS * SDIM];          // 128   KB score stripe
  __shared__ unsigned Hist[NWAVES * 256];       //  8    KB radix bins

  const int bh = blockIdx.x >> 7;               // 128 t-tiles per (b,h)
  const int t0 = (blockIdx.x & 127) * ROWS;
  const size_t rowbase = (size_t)bh * SDIM;
  const int bucket = *bucket_p;

  // Stage A tile (16 x 66, zero-padded to 68) into LDS
  for (int i = threadIdx.x; i < ROWS * KDIM; i += 256) {
    int mm = i / KDIM, k = i % KDIM;
    Alds[i] = (k < ADIM) ? a[(rowbase + t0 + mm) * ADIM + k] : 0.0f;
  }
  __syncthreads();

  const int wave  = threadIdx.x >> 5;
  const int lane  = threadIdx.x & 31;
  const int l16   = lane & 15;
  const int khalf = (lane >> 4) * 2;            // K sub-offset per lane half

  // ---- GEMM: each wave owns N-tiles wave, wave+8, ... (16 tiles of 16 cols)
  for (int nt = wave; nt < SDIM / 16; nt += NWAVES) {
    const int s0 = nt * 16;
    v8f acc = {};
    const float* prow = pq + (rowbase + s0 + l16) * KDIM;   // B row = Pq[s0+n]
    #pragma unroll
    for (int k0 = 0; k0 < KDIM; k0 += 4) {
      v2f af, bf;
      af.x = Alds[l16 * KDIM + k0 + khalf];     // A[m=l16, k0+khalf .. +1]
      af.y = Alds[l16 * KDIM + k0 + khalf + 1];
      bf.x = prow[k0 + khalf];                  // B[k, n=l16] = Pq[s0+n, k]
      bf.y = prow[k0 + khalf + 1];
      acc = __builtin_amdgcn_wmma_f32_16x16x4_f32(
          false, af, false, bf, (short)0, acc, false, false);
    }
    // Spill 16x16 tile to LDS stripe, zeroing NaNs (reference semantics)
    const int mbase = (lane >> 4) * 8;
    const int n = s0 + l16;
    #pragma unroll
    for (int r = 0; r < 8; ++r) {
      float v = acc[r];
      v = (v == v) ? v : 0.0f;
      Sc[(mbase + r) * SDIM + n] = v;
    }
  }
  __syncthreads();

  // ---- Top-k per row: wave w handles rows 2w, 2w+1. Row cached once in
  // registers as monotone keys; 4-pass radix select finds k-th largest key;
  // mask written straight from registers (coalesced b128 stores).
  unsigned* hist = &Hist[wave * 256];
  for (int rr = 0; rr < 2; ++rr) {
    const int m = wave * 2 + rr;
    const uint4* rowq = (const uint4*)&Sc[(size_t)m * SDIM];

    uint4 U[16];                       // element (j, lane, c) has index
    #pragma unroll                     // g = j*128 + lane*4 + c
    for (int j = 0; j < 16; ++j) {
      uint4 raw = rowq[j * 32 + lane];
      U[j].x = mkey(raw.x); U[j].y = mkey(raw.y);
      U[j].z = mkey(raw.z); U[j].w = mkey(raw.w);
    }

    // radix select: find key of k-th largest + how many ties to keep
    unsigned kk = (unsigned)bucket;
    unsigned prefix = 0, pmask = 0, eqcnt = 0;
    #pragma unroll
    for (int pass = 0; pass < 4; ++pass) {
      const int shift = 24 - 8 * pass;
      #pragma unroll
      for (int b = 0; b < 8; ++b) hist[lane * 8 + b] = 0u;     // zero bins
      #pragma unroll
      for (int j = 0; j < 16; ++j) {                           // histogram
        unsigned uu;
        uu = U[j].x; if ((uu & pmask) == prefix) atomicAdd(&hist[(uu >> shift) & 255], 1u);
        uu = U[j].y; if ((uu & pmask) == prefix) atomicAdd(&hist[(uu >> shift) & 255], 1u);
        uu = U[j].z; if ((uu & pmask) == prefix) atomicAdd(&hist[(uu >> shift) & 255], 1u);
        uu = U[j].w; if ((uu & pmask) == prefix) atomicAdd(&hist[(uu >> shift) & 255], 1u);
      }
      unsigned h[8], s = 0;                                    // lane's 8 bins
      #pragma unroll
      for (int b = 0; b < 8; ++b) { h[b] = hist[lane * 8 + b]; s += h[b]; }
      unsigned t = s;                                          // suffix scan
      #pragma unroll
      for (int off = 1; off < 32; off <<= 1) {
        unsigned o = __shfl_down(t, off, 32);
        if (lane + off < 32) t += o;
      }
      unsigned C = t - s;              // count of elems in bins above lane's
      unsigned long long ball = __ballot(C < kk && t >= kk);
      int lstar = __ffsll(ball) - 1;   // unique lane holding target bin
      unsigned digit = 0, kknew = 0, eql = 0;
      if (lane == lstar) {
        unsigned c = C;
        #pragma unroll
        for (int b = 7; b >= 0; --b) { // high digit first within lane
          if (c + h[b] >= kk) { digit = (unsigned)(lane * 8 + b); kknew = kk - c; eql = h[b]; break; }
          c += h[b];
        }
      }
      digit = __shfl((int)digit, lstar, 32);
      kk    = __shfl((int)kknew, lstar, 32);
      eqcnt = __shfl((int)eql,   lstar, 32);
      prefix |= digit << shift;
      pmask  |= 0xFFu << shift;
    }
    const unsigned tkey = prefix;      // exact key of k-th largest element

    // mask write: 0 for selected, -10000 otherwise
    const size_t obase = (rowbase + t0 + m) * (size_t)SDIM;
    float4* orow = (float4*)(out + obase);
    if (kk == eqcnt) {
      // no boundary ties: select exactly u >= tkey  (common case)
      #pragma unroll
      for (int j = 0; j < 16; ++j) {
        float4 vv;
        vv.x = (U[j].x >= tkey) ? 0.0f : -10000.0f;
        vv.y = (U[j].y >= tkey) ? 0.0f : -10000.0f;
        vv.z = (U[j].z >= tkey) ? 0.0f : -10000.0f;
        vv.w = (U[j].w >= tkey) ? 0.0f : -10000.0f;
        orow[j * 32 + lane] = vv;
      }
    } else {
      // boundary ties: keep the kk lowest-indexed elements equal to tkey
      unsigned base = 0;
      for (int j = 0; j < 16; ++j) {
        unsigned e0 = (U[j].x == tkey), e1 = (U[j].y == tkey);
        unsigned e2 = (U[j].z == tkey), e3 = (U[j].w == tkey);
        unsigned en = e0 + e1 + e2 + e3;
        unsigned sc = en;                       // inclusive scan over lanes
        #pragma unroll
        for (int off = 1; off < 32; off <<= 1) {
          unsigned o = __shfl_up(sc, off, 32);
          if (lane >= off) sc += o;
        }
        unsigned r0 = base + sc - en;           // rank of this lane's first eq
        unsigned r1 = r0 + e0, r2 = r1 + e1, r3 = r2 + e2;
        float4 vv;
        vv.x = (U[j].x > tkey || (e0 && r0 < kk)) ? 0.0f : -10000.0f;
        vv.y = (U[j].y > tkey || (e1 && r1 < kk)) ? 0.0f : -10000.0f;
        vv.z = (U[j].z > tkey || (e2 && r2 < kk)) ? 0.0f : -10000.0f;
        vv.w = (U[j].w > tkey || (e3 && r3 < kk)) ? 0.0f : -10000.0f;
        orow[j * 32 + lane] = vv;
        base += (unsigned)__shfl((int)sc, 31, 32);
      }
    }
  }
}

// ---------------------------------------------------------------------------
extern "C" void kernel_launch(void* const* d_in, const int* in_sizes, int n_in,
                              void* d_out, int out_size, void* d_ws, size_t ws_size,
                              hipStream_t stream) {
  const float* qk     = (const float*)d_in[0];   // [B,H,S,64]
  const float* a      = (const float*)d_in[1];   // [B,H,S,66]
  const int*   bucket = (const int*)d_in[2];     // scalar (32)
  float*       out    = (float*)d_out;           // [B,H,S,S]

  const int nrows = in_sizes[0] / DDIM;          // B*H*S = 32768
  const int nbh   = nrows / SDIM;                // B*H   = 16

  unsigned* wsmax = (unsigned*)d_ws;
  float*    pq    = (float*)((char*)d_ws + 256); // nrows * 68 floats (~8.9 MB)

  hipMemsetAsync(wsmax, 0, sizeof(unsigned), stream);

  dim3 blk(256);
  dim3 grid_rows((nrows + NWAVES - 1) / NWAVES);
  k_maxnorm <<<grid_rows, blk, 0, stream>>>(qk, wsmax, nrows);
  k_build_pq<<<grid_rows, blk, 0, stream>>>(qk, a, wsmax, pq, nrows);

  dim3 grid_tiles(nbh * (SDIM / ROWS));          // 16 * 128 = 2048 WGs
  k_scores_topk<<<grid_tiles, blk, 0, stream>>>(a, pq, bucket, out);
}